// VGAE_48584670052621
// MI455X (gfx1250) — compile-verified
//
#include <hip/hip_runtime.h>

typedef float v2f __attribute__((ext_vector_type(2)));
typedef float v8f __attribute__((ext_vector_type(8)));

// ---------------------------------------------------------------------------
// Elementwise helpers
// ---------------------------------------------------------------------------
__global__ __launch_bounds__(256) void vgae_fill_zero(float* __restrict__ p, long long n) {
  long long i = (long long)blockIdx.x * blockDim.x + threadIdx.x;
  if (i < n) p[i] = 0.0f;
}

__global__ __launch_bounds__(256) void vgae_relu_inplace(float* __restrict__ p, long long n) {
  long long i = (long long)blockIdx.x * blockDim.x + threadIdx.x;
  if (i < n) p[i] = fmaxf(p[i], 0.0f);
}

__global__ __launch_bounds__(256) void vgae_exp(const float* __restrict__ in,
                                                float* __restrict__ out, long long n) {
  long long i = (long long)blockIdx.x * blockDim.x + threadIdx.x;
  if (i < n) out[i] = __expf(in[i]);
}

// ---------------------------------------------------------------------------
// SpMM as edge scatter: out[dst] += w * feat[src]  (float4 per thread)
// ---------------------------------------------------------------------------
__global__ __launch_bounds__(256) void vgae_spmm_scatter(
    const float* __restrict__ feat, const int* __restrict__ esrc,
    const int* __restrict__ edst, const float* __restrict__ ew,
    float* __restrict__ out, int E, int F) {
  const int groups = F >> 2;  // float4 groups per row
  long long tid = (long long)blockIdx.x * blockDim.x + threadIdx.x;
  long long total = (long long)E * groups;
  if (tid >= total) return;
  int e = (int)(tid / groups);
  int g = (int)(tid % groups) << 2;
  int s = esrc[e];
  int d = edst[e];
  float w = ew[e];
  const float4 v = *(const float4*)(feat + (long long)s * F + g);
  float* o = out + (long long)d * F + g;
  atomicAdd(o + 0, v.x * w);
  atomicAdd(o + 1, v.y * w);
  atomicAdd(o + 2, v.z * w);
  atomicAdd(o + 3, v.w * w);
}

// ---------------------------------------------------------------------------
// Generic FP32 WMMA GEMM: C[M,N] = A[M,K] @ B[K,N].
// One wave per 16x16 tile, V_WMMA_F32_16X16X4_F32, K-step of 4.
// Used for the small GEMMs (x@W1, h@W2_*). Wave-uniform tile guard keeps
// EXEC all-ones inside WMMA (ISA 7.12 requirement).
// ---------------------------------------------------------------------------
__global__ __launch_bounds__(256) void vgae_wmma_gemm_f32(
    const float* __restrict__ A, const float* __restrict__ B,
    float* __restrict__ C, int M, int N, int K) {
  const int wave = (int)(((long long)blockIdx.x * blockDim.x + threadIdx.x) >> 5);
  const int lane = threadIdx.x & 31;
  const int tilesN = N >> 4;
  const int tilesM = M >> 4;
  if (wave >= tilesM * tilesN) return;

  const int tm = wave / tilesN;
  const int tn = wave % tilesN;
  const int row0 = tm << 4;
  const int col0 = tn << 4;
  const int hl = lane & 15;           // element index within half-wave
  const int koff = (lane >> 4) << 1;  // lanes 16-31 hold K+2, K+3

  v8f c = {0.f, 0.f, 0.f, 0.f, 0.f, 0.f, 0.f, 0.f};

  const float* aRow = A + (long long)(row0 + hl) * K;

  for (int k = 0; k < K; k += 4) {
    v2f a = *(const v2f*)(aRow + k + koff);
    v2f b;
    b.x = B[(long long)(k + koff) * N + col0 + hl];
    b.y = B[(long long)(k + koff + 1) * N + col0 + hl];
    c = __builtin_amdgcn_wmma_f32_16x16x4_f32(false, a, false, b,
                                              (short)0, c, false, false);
  }

  float* cPtr = C + (long long)(row0 + ((lane >> 4) << 3)) * N + col0 + hl;
#pragma unroll
  for (int r = 0; r < 8; ++r) {
    cPtr[(long long)r * N] = c[r];
  }
}

// ---------------------------------------------------------------------------
// adj = mu @ mu^T specialization (M x M output, K = 64).
// K is small enough that the entire 16x64 A fragment fits in 16 v2f regs
// per lane (32 VGPRs). Each wave loads A once and reuses it across CT=5
// column tiles (16x80 output strip): per tile, 16 b64 B loads feed 16
// back-to-back WMMAs. Drops operand traffic ~5x vs the generic kernel on
// the 12.8 GFLOP dominant op; mu (2.5 MB) stays L2-resident.
// ---------------------------------------------------------------------------
__global__ __launch_bounds__(256) void vgae_wmma_adj(
    const float* __restrict__ mu, float* __restrict__ C, int M) {
  constexpr int K = 64;
  constexpr int CT = 5;  // column tiles per wave (625 = 5 * 125)
  const int wave = (int)(((long long)blockIdx.x * blockDim.x + threadIdx.x) >> 5);
  const int lane = threadIdx.x & 31;
  const int tilesM = M >> 4;        // 625
  const int groupsN = tilesM / CT;  // 125
  if (wave >= tilesM * groupsN) return;

  const int tm = wave / groupsN;
  const int tg = wave % groupsN;
  const int row0 = tm << 4;
  const int col0 = (tg * CT) << 4;
  const int hl = lane & 15;
  const int koff = (lane >> 4) << 1;

  // Load full A fragment for this 16-row strip: 16 k-steps x v2f.
  const float* aRow = mu + (long long)(row0 + hl) * K + koff;
  v2f a[K / 4];
#pragma unroll
  for (int kk = 0; kk < K / 4; ++kk) {
    a[kk] = *(const v2f*)(aRow + 4 * kk);
  }

  for (int ct = 0; ct < CT; ++ct) {
    const int col = col0 + (ct << 4);
    // B^T[k][col+hl] = mu[col+hl][k] -> contiguous along K per lane.
    const float* bRow = mu + (long long)(col + hl) * K + koff;

    // Prefetch next tile's B row into cache (global_prefetch_b8).
    if (ct + 1 < CT) {
      __builtin_prefetch(bRow + ((long long)16 * K), 0, 0);
    }

    v2f b[K / 4];
#pragma unroll
    for (int kk = 0; kk < K / 4; ++kk) {
      b[kk] = *(const v2f*)(bRow + 4 * kk);
    }

    v8f c = {0.f, 0.f, 0.f, 0.f, 0.f, 0.f, 0.f, 0.f};
#pragma unroll
    for (int kk = 0; kk < K / 4; ++kk) {
      c = __builtin_amdgcn_wmma_f32_16x16x4_f32(false, a[kk], false, b[kk],
                                                (short)0, c, false, false);
    }

    float* cPtr = C + (long long)(row0 + ((lane >> 4) << 3)) * M + col + hl;
#pragma unroll
    for (int r = 0; r < 8; ++r) {
      cPtr[(long long)r * M] = c[r];
    }
  }
}

// ---------------------------------------------------------------------------
// Launcher
// ---------------------------------------------------------------------------
extern "C" void kernel_launch(void* const* d_in, const int* in_sizes, int n_in,
                              void* d_out, int out_size, void* d_ws, size_t ws_size,
                              hipStream_t stream) {
  const float* x     = (const float*)d_in[0];  // [M, D]
  const int*   esrc  = (const int*)d_in[1];    // [E]
  const int*   edst  = (const int*)d_in[2];    // [E]
  const float* ew    = (const float*)d_in[3];  // [E]
  const float* W1    = (const float*)d_in[4];  // [D, H]
  const float* W2mu  = (const float*)d_in[5];  // [H, L]
  const float* W2sig = (const float*)d_in[6];  // [H, L]

  const int D = 512, H = 256, L = 64;
  const int M = in_sizes[0] / D;   // 10000 nodes
  const int E = in_sizes[1];       // 320000 edges

  // Workspace layout (floats)
  float* ws     = (float*)d_ws;
  float* xw1    = ws;                               // M*H
  float* h      = xw1 + (long long)M * H;           // M*H
  float* hwmu   = h + (long long)M * H;             // M*L
  float* hwsig  = hwmu + (long long)M * L;          // M*L
  float* sigacc = hwsig + (long long)M * L;         // M*L

  // Output layout: mu [M*L] | sigma [M*L] | adj [M*M]
  float* out_mu  = (float*)d_out;
  float* out_sig = out_mu + (long long)M * L;
  float* out_adj = out_sig + (long long)M * L;

  auto blocks = [](long long n) { return (unsigned)((n + 255) / 256); };
  auto wave_blocks = [](long long waves) {
    return (unsigned)((waves * 32 + 255) / 256);
  };

  // 1) Zero the atomic accumulators (h, mu-out, sigma-acc)
  vgae_fill_zero<<<blocks((long long)M * H), 256, 0, stream>>>(h, (long long)M * H);
  vgae_fill_zero<<<blocks((long long)M * L), 256, 0, stream>>>(out_mu, (long long)M * L);
  vgae_fill_zero<<<blocks((long long)M * L), 256, 0, stream>>>(sigacc, (long long)M * L);

  // 2) xw1 = x @ W1   (M x D x H)
  vgae_wmma_gemm_f32<<<wave_blocks((long long)(M / 16) * (H / 16)), 256, 0, stream>>>(
      x, W1, xw1, M, H, D);

  // 3) h = spmm(xw1); then ReLU in place
  vgae_spmm_scatter<<<blocks((long long)E * (H >> 2)), 256, 0, stream>>>(
      xw1, esrc, edst, ew, h, E, H);
  vgae_relu_inplace<<<blocks((long long)M * H), 256, 0, stream>>>(h, (long long)M * H);

  // 4) head GEMMs: hwmu = h @ W2_mu, hwsig = h @ W2_sig   (M x H x L)
  vgae_wmma_gemm_f32<<<wave_blocks((long long)(M / 16) * (L / 16)), 256, 0, stream>>>(
      h, W2mu, hwmu, M, L, H);
  vgae_wmma_gemm_f32<<<wave_blocks((long long)(M / 16) * (L / 16)), 256, 0, stream>>>(
      h, W2sig, hwsig, M, L, H);

  // 5) mu = spmm(hwmu) -> d_out directly; sigacc = spmm(hwsig)
  vgae_spmm_scatter<<<blocks((long long)E * (L >> 2)), 256, 0, stream>>>(
      hwmu, esrc, edst, ew, out_mu, E, L);
  vgae_spmm_scatter<<<blocks((long long)E * (L >> 2)), 256, 0, stream>>>(
      hwsig, esrc, edst, ew, sigacc, E, L);

  // 6) sigma = exp(sigacc)
  vgae_exp<<<blocks((long long)M * L), 256, 0, stream>>>(sigacc, out_sig, (long long)M * L);

  // 7) adj = mu @ mu^T, register-blocked A reuse (16x80 strip per wave)
  {
    long long waves = (long long)(M / 16) * ((M / 16) / 5);
    vgae_wmma_adj<<<wave_blocks(waves), 256, 0, stream>>>(out_mu, out_adj, M);
  }
}